// GaussianMixtureUnif_basic_75797582840307
// MI455X (gfx1250) — compile-verified
//
#include <hip/hip_runtime.h>
#include <math.h>

// ---------------------------------------------------------------------------
// GMM EM step for MI455X (gfx1250, wave32).
// x . mu^T dot products computed with V_WMMA_F32_16X16X4_F32 (K=4 pads D=3).
// Fragment builds use unconditional LDS loads + selects (no exec-mask branches).
// ---------------------------------------------------------------------------

typedef float v2f __attribute__((ext_vector_type(2)));
typedef float v8f __attribute__((ext_vector_type(8)));

#define GMM_C    256   // clusters (fixed by reference)
#define ROWS     32    // rows of X per block
#define BTHREADS 64    // 2 waves of 32

// workspace layout (float offsets)
enum {
  WS_ZW = 0,      // logsumexp(w)
  WS_SSUM,        // sum gamma * D2
  WS_LGN,         // d*(log sigma + 0.5 log 2pi)  (old sigma)
  WS_INV2S2,      // 1/(2 sigma^2)                (old sigma)
  WS_TWOS2,       // 2 sigma^2                    (old sigma)
  WS_INV2S2N,     // 1/(2 sigma_new^2)
  WS_S = 8,                        // [256]  S_c = sum_n gamma
  WS_MNUM  = WS_S + GMM_C,         // [768]  sum_n gamma * X
  WS_MUNEW = WS_MNUM + 3 * GMM_C,  // [768]  mu_new
  WS_MU2N  = WS_MUNEW + 3 * GMM_C, // [256]  ||mu_new||^2
  WS_LPI   = WS_MU2N + GMM_C       // [256]  log pi_c
};

// ---------------------------------------------------------------------------
// Kernel 0: zero accumulators, Zw = logsumexp(w), fold sigma constants.
// ---------------------------------------------------------------------------
__global__ __launch_bounds__(GMM_C)
void gmm_init_kernel(const float* __restrict__ w, const float* __restrict__ sigma,
                     float* __restrict__ ws) {
  __shared__ float red[GMM_C];
  const int t = threadIdx.x;
  ws[WS_S + t] = 0.0f;
  ws[WS_MNUM + 3 * t + 0] = 0.0f;
  ws[WS_MNUM + 3 * t + 1] = 0.0f;
  ws[WS_MNUM + 3 * t + 2] = 0.0f;

  const float v = w[t];
  red[t] = v;
  __syncthreads();
  for (int s = GMM_C / 2; s > 0; s >>= 1) {
    if (t < s) red[t] = fmaxf(red[t], red[t + s]);
    __syncthreads();
  }
  const float m = red[0];
  __syncthreads();
  red[t] = __expf(v - m);
  __syncthreads();
  for (int s = GMM_C / 2; s > 0; s >>= 1) {
    if (t < s) red[t] += red[t + s];
    __syncthreads();
  }
  if (t == 0) {
    const float Zw  = m + __logf(red[0]);
    const float sig = sigma[0];
    ws[WS_ZW]     = Zw;
    ws[WS_LGN]    = 3.0f * (__logf(sig) + 0.5f * __logf(6.283185307179586f));
    ws[WS_INV2S2] = 1.0f / (2.0f * sig * sig);
    ws[WS_TWOS2]  = 2.0f * sig * sig;
    ws[WS_SSUM]   = 0.0f;
  }
}

// ---------------------------------------------------------------------------
// Shared device helper: compute the 32x256 tile of t_nc into LDS via WMMA.
// t_nc = (w_c - Zw - lgn) - (||x||^2 + ||mu||^2 - 2 x.mu) / (2 sigma^2)
// A fragment (16x4 f32): lanes 0-15 carry K=0,1 ; lanes 16-31 carry K=2,3.
// B fragment (4x16 f32): same split; K=3 is zero padding (D=3).
// All selects read pre-loaded temps -> v_cndmask, no exec-masked loads.
// ---------------------------------------------------------------------------
__device__ __forceinline__
void gmm_t_tile_wmma(float* tT, const float* Xs, const float* x2S,
                     const float* muXs, const float* muYs, const float* muZs,
                     const float* mu2S, const float* wcS,
                     float inv2s2, int tid) {
  const int wave  = tid >> 5;
  const int lane  = tid & 31;
  const int half  = lane >> 4;
  const int l     = lane & 15;
  const int rbase = wave * 16;

  const float xa0 = Xs[3 * (rbase + l) + 0];
  const float xa1 = Xs[3 * (rbase + l) + 1];
  const float xa2 = Xs[3 * (rbase + l) + 2];
  v2f a;
  a.x = half ? xa2  : xa0;
  a.y = half ? 0.0f : xa1;

  for (int ct = 0; ct < GMM_C / 16; ++ct) {
    const int c = ct * 16 + l;
    const float mb0 = muXs[c];
    const float mb1 = muYs[c];
    const float mb2 = muZs[c];
    v2f b;
    b.x = half ? mb2  : mb0;
    b.y = half ? 0.0f : mb1;
    v8f cz = {};
    v8f d = __builtin_amdgcn_wmma_f32_16x16x4_f32(
        /*neg_a=*/false, a, /*neg_b=*/false, b,
        /*c_mod=*/(short)0, cz, /*reuse_a=*/false, /*reuse_b=*/false);
#pragma unroll
    for (int r = 0; r < 8; ++r) {
      const int rowL = rbase + r + 8 * half;            // M index of d[r]
      const float tv =
          wcS[c] - (x2S[rowL] + mu2S[c] - 2.0f * d[r]) * inv2s2;
      tT[rowL * GMM_C + c] = tv;
    }
  }
}

// row-wise logsumexp over the LDS tile; 2 threads per row, 128 c's each.
__device__ __forceinline__
void gmm_row_logsumexp(const float* tT, float* Ts, float* red, int tid) {
  const int row  = tid >> 1;
  const int part = tid & 1;
  const float* tp = &tT[row * GMM_C + part * 128];
  float m = -1e30f;
  for (int j = 0; j < 128; ++j) m = fmaxf(m, tp[(j + row) & 127]);
  red[tid] = m;
  __syncthreads();
  const float M = fmaxf(red[2 * row], red[2 * row + 1]);
  __syncthreads();
  float s = 0.0f;
  for (int j = 0; j < 128; ++j) s += __expf(tp[(j + row) & 127] - M);
  red[tid] = s;
  __syncthreads();
  if (part == 0) Ts[row] = M + __logf(red[tid] + red[tid + 1]);
  __syncthreads();
}

// ---------------------------------------------------------------------------
// Kernel 1: E-step + M-step accumulators (S_c, sum gamma*X, sum gamma*D2).
// ---------------------------------------------------------------------------
__global__ __launch_bounds__(BTHREADS)
void gmm_estep_kernel(const float* __restrict__ X, const float* __restrict__ mu,
                      const float* __restrict__ w, float* __restrict__ ws, int n) {
  __shared__ float tT[ROWS * GMM_C];   // 32 KB
  __shared__ float muXs[GMM_C];
  __shared__ float muYs[GMM_C];
  __shared__ float muZs[GMM_C];
  __shared__ float mu2S[GMM_C];
  __shared__ float wcS[GMM_C];
  __shared__ float Xs[3 * ROWS];
  __shared__ float x2S[ROWS];
  __shared__ float Ts[ROWS];
  __shared__ float red[BTHREADS];

  const int tid = threadIdx.x;
  const int n0  = blockIdx.x * ROWS;
  const float Zw     = ws[WS_ZW];
  const float lgn    = ws[WS_LGN];
  const float inv2s2 = ws[WS_INV2S2];
  const float twoS2  = ws[WS_TWOS2];

  for (int c = tid; c < GMM_C; c += BTHREADS) {
    const float m0 = mu[3 * c + 0], m1 = mu[3 * c + 1], m2 = mu[3 * c + 2];
    muXs[c] = m0; muYs[c] = m1; muZs[c] = m2;
    mu2S[c] = m0 * m0 + m1 * m1 + m2 * m2;
    wcS[c]  = w[c] - Zw - lgn;
  }
  if (tid < ROWS) {
    int r = n0 + tid; if (r > n - 1) r = n - 1;
    const float xa = X[3 * r + 0], xb = X[3 * r + 1], xc = X[3 * r + 2];
    Xs[3 * tid + 0] = xa; Xs[3 * tid + 1] = xb; Xs[3 * tid + 2] = xc;
    x2S[tid] = xa * xa + xb * xb + xc * xc;
  }
  if (tid == 0 && n0 + ROWS < n) {
    __builtin_prefetch(&X[3 * (n0 + ROWS)], 0, 1);   // global_prefetch_b8
  }
  __syncthreads();

  gmm_t_tile_wmma(tT, Xs, x2S, muXs, muYs, muZs, mu2S, wcS, inv2s2, tid);
  __syncthreads();

  gmm_row_logsumexp(tT, Ts, red, tid);

  // per-cluster accumulation: thread <-> column, conflict-free LDS reads
  const int rmax = (n - n0 < ROWS) ? (n - n0) : ROWS;
  float ssum_local = 0.0f;
  for (int cc = tid; cc < GMM_C; cc += BTHREADS) {
    float sC = 0.0f, a0 = 0.0f, a1 = 0.0f, a2 = 0.0f;
    const float wc = wcS[cc];
    for (int row = 0; row < rmax; ++row) {
      const float tv = tT[row * GMM_C + cc];
      const float g  = __expf(tv - Ts[row]);
      sC += g;
      a0 += g * Xs[3 * row + 0];
      a1 += g * Xs[3 * row + 1];
      a2 += g * Xs[3 * row + 2];
      ssum_local += g * (wc - tv);      // = gamma * D2 / (2 sigma^2)
    }
    atomicAdd(&ws[WS_S + cc], sC);
    atomicAdd(&ws[WS_MNUM + 3 * cc + 0], a0);
    atomicAdd(&ws[WS_MNUM + 3 * cc + 1], a1);
    atomicAdd(&ws[WS_MNUM + 3 * cc + 2], a2);
  }
  ssum_local *= twoS2;                  // -> gamma * D2
  red[tid] = ssum_local;
  __syncthreads();
  for (int s = BTHREADS / 2; s > 0; s >>= 1) {
    if (tid < s) red[tid] += red[tid + s];
    __syncthreads();
  }
  if (tid == 0) atomicAdd(&ws[WS_SSUM], red[0]);
}

// ---------------------------------------------------------------------------
// Kernel 2: finalize cluster stats (mu_new, w_new -> lpi, sigma_new).
// ---------------------------------------------------------------------------
__global__ __launch_bounds__(GMM_C)
void gmm_mstep_kernel(float* __restrict__ ws, float* __restrict__ cfe_out, int n) {
  __shared__ float red[GMM_C];
  const int c = threadIdx.x;
  const float S = ws[WS_S + c];
  red[c] = S;
  __syncthreads();
  for (int s = GMM_C / 2; s > 0; s >>= 1) {
    if (c < s) red[c] += red[c + s];
    __syncthreads();
  }
  const float sumS = red[0];
  const float invS = 1.0f / S;
  const float m0 = ws[WS_MNUM + 3 * c + 0] * invS;
  const float m1 = ws[WS_MNUM + 3 * c + 1] * invS;
  const float m2 = ws[WS_MNUM + 3 * c + 2] * invS;
  ws[WS_MUNEW + 3 * c + 0] = m0;
  ws[WS_MUNEW + 3 * c + 1] = m1;
  ws[WS_MUNEW + 3 * c + 2] = m2;
  ws[WS_MU2N + c] = m0 * m0 + m1 * m1 + m2 * m2;
  ws[WS_LPI + c]  = __logf(S) - __logf(sumS);   // w_new - logsumexp(w_new)
  if (c == 0) {
    const float s2 = ws[WS_SSUM] / (3.0f * (float)n);   // sigma_new^2
    ws[WS_INV2S2N] = 1.0f / (2.0f * s2);
    cfe_out[0] = 0.0f;                                   // zero Cfe accumulator
  }
}

// ---------------------------------------------------------------------------
// Kernel 3: recompute t (WMMA) + T_n, emit Y = gamma @ mu_new and Cfe.
// Uses sum_c gamma = 1 to fold the (-||Y||^2/(2 s^2) + lgn) terms per row.
// ---------------------------------------------------------------------------
__global__ __launch_bounds__(BTHREADS)
void gmm_output_kernel(const float* __restrict__ X, const float* __restrict__ mu,
                       const float* __restrict__ w, const float* __restrict__ ws,
                       float* __restrict__ out, float* __restrict__ cfe_out, int n) {
  __shared__ float tT[ROWS * GMM_C];
  __shared__ float muXs[GMM_C];
  __shared__ float muYs[GMM_C];
  __shared__ float muZs[GMM_C];
  __shared__ float mu2S[GMM_C];
  __shared__ float wcS[GMM_C];
  __shared__ float munS[3 * GMM_C];
  __shared__ float mu2nS[GMM_C];
  __shared__ float lpiS[GMM_C];
  __shared__ float Xs[3 * ROWS];
  __shared__ float x2S[ROWS];
  __shared__ float Ts[ROWS];
  __shared__ float pr[BTHREADS * 4];
  __shared__ float red[BTHREADS];

  const int tid = threadIdx.x;
  const int n0  = blockIdx.x * ROWS;
  const float Zw      = ws[WS_ZW];
  const float lgn     = ws[WS_LGN];
  const float inv2s2  = ws[WS_INV2S2];
  const float inv2s2n = ws[WS_INV2S2N];

  for (int c = tid; c < GMM_C; c += BTHREADS) {
    const float m0 = mu[3 * c + 0], m1 = mu[3 * c + 1], m2 = mu[3 * c + 2];
    muXs[c] = m0; muYs[c] = m1; muZs[c] = m2;
    mu2S[c] = m0 * m0 + m1 * m1 + m2 * m2;
    wcS[c]  = w[c] - Zw - lgn;
    munS[3 * c + 0] = ws[WS_MUNEW + 3 * c + 0];
    munS[3 * c + 1] = ws[WS_MUNEW + 3 * c + 1];
    munS[3 * c + 2] = ws[WS_MUNEW + 3 * c + 2];
    mu2nS[c] = ws[WS_MU2N + c];
    lpiS[c]  = ws[WS_LPI + c];
  }
  if (tid < ROWS) {
    int r = n0 + tid; if (r > n - 1) r = n - 1;
    const float xa = X[3 * r + 0], xb = X[3 * r + 1], xc = X[3 * r + 2];
    Xs[3 * tid + 0] = xa; Xs[3 * tid + 1] = xb; Xs[3 * tid + 2] = xc;
    x2S[tid] = xa * xa + xb * xb + xc * xc;
  }
  if (tid == 0 && n0 + ROWS < n) {
    __builtin_prefetch(&X[3 * (n0 + ROWS)], 0, 1);
  }
  __syncthreads();

  gmm_t_tile_wmma(tT, Xs, x2S, muXs, muYs, muZs, mu2S, wcS, inv2s2, tid);
  __syncthreads();

  gmm_row_logsumexp(tT, Ts, red, tid);   // identical recompute of T_n

  const int row  = tid >> 1;
  const int part = tid & 1;
  const int gr   = n0 + row;
  const float Tn = Ts[row];
  float y0 = 0.0f, y1 = 0.0f, y2 = 0.0f, acc = 0.0f;
  for (int j = 0; j < 128; ++j) {
    const int c = part * 128 + ((j + row) & 127);
    const float tv = tT[row * GMM_C + c];
    const float lg = tv - Tn;
    const float g  = __expf(lg);
    y0  += g * munS[3 * c + 0];
    y1  += g * munS[3 * c + 1];
    y2  += g * munS[3 * c + 2];
    acc += g * (mu2nS[c] * inv2s2n + lg - lpiS[c]);
  }
  pr[4 * tid + 0] = y0; pr[4 * tid + 1] = y1;
  pr[4 * tid + 2] = y2; pr[4 * tid + 3] = acc;
  __syncthreads();

  float cfe = 0.0f;
  if (part == 0) {
    const float Y0 = pr[4 * tid + 0] + pr[4 * tid + 4];
    const float Y1 = pr[4 * tid + 1] + pr[4 * tid + 5];
    const float Y2 = pr[4 * tid + 2] + pr[4 * tid + 6];
    const float A  = pr[4 * tid + 3] + pr[4 * tid + 7];
    if (gr < n) {
      out[3 * gr + 0] = Y0; out[3 * gr + 1] = Y1; out[3 * gr + 2] = Y2;
      const float yn2 = Y0 * Y0 + Y1 * Y1 + Y2 * Y2;
      cfe = A - yn2 * inv2s2n + lgn;
    }
  }
  red[tid] = cfe;
  __syncthreads();
  for (int s = BTHREADS / 2; s > 0; s >>= 1) {
    if (tid < s) red[tid] += red[tid + s];
    __syncthreads();
  }
  if (tid == 0) atomicAdd(cfe_out, red[0]);
}

// ---------------------------------------------------------------------------
// Host launcher
// ---------------------------------------------------------------------------
extern "C" void kernel_launch(void* const* d_in, const int* in_sizes, int n_in,
                              void* d_out, int out_size, void* d_ws, size_t ws_size,
                              hipStream_t stream) {
  const float* X     = (const float*)d_in[0];
  const float* mu    = (const float*)d_in[1];
  const float* w     = (const float*)d_in[2];
  const float* sigma = (const float*)d_in[3];
  const int n = in_sizes[0] / 3;

  float* ws  = (float*)d_ws;
  float* out = (float*)d_out;
  float* cfe = out + (size_t)n * 3;   // Cfe scalar follows Y in d_out

  gmm_init_kernel<<<1, GMM_C, 0, stream>>>(w, sigma, ws);
  const int nb = (n + ROWS - 1) / ROWS;
  gmm_estep_kernel<<<nb, BTHREADS, 0, stream>>>(X, mu, w, ws, n);
  gmm_mstep_kernel<<<1, GMM_C, 0, stream>>>(ws, cfe, n);
  gmm_output_kernel<<<nb, BTHREADS, 0, stream>>>(X, mu, w, ws, out, cfe, n);
}